// AnatomicalContrastiveLoss_85813446574305
// MI455X (gfx1250) — compile-verified
//
#include <hip/hip_runtime.h>
#include <hip/hip_bf16.h>

#define B_ 4
#define C_ 4
#define V_ 262144
#define F_ 64
#define K_ 100

typedef float v2f __attribute__((ext_vector_type(2)));
typedef float v8f __attribute__((ext_vector_type(8)));

#define VB 64      // v-chunk per block iteration
#define EPAD 68    // LDS row stride in elements (+4 pad: conflict-free, 16B-aligned rows)
#define NCHUNK (V_ / VB)

// ---------------------------------------------------------------------------
// CDNA5 async global->LDS copy (ASYNCcnt-tracked), via inline asm.
// vdst = LDS byte address (low 32 bits of generic pointer), vaddr = 64-bit VA.
// ---------------------------------------------------------------------------
__device__ __forceinline__ void async_copy_b128(void* lds_ptr, const void* gptr) {
    unsigned int lds_off = (unsigned int)(unsigned long long)lds_ptr;
    asm volatile("global_load_async_to_lds_b128 %0, %1, off"
                 :: "v"(lds_off), "v"(gptr) : "memory");
}
__device__ __forceinline__ void wait_async_zero() {
    asm volatile("s_wait_asynccnt 0x0" ::: "memory");
}

// ---------------------------------------------------------------------------
// Kernel 0: zero the class_sum / count accumulators
// ---------------------------------------------------------------------------
__global__ void init_ws_kernel(float* cs, float* cnt) {
    int t = threadIdx.x;
    if (t < C_ * F_) cs[t] = 0.0f;
    if (t < C_)      cnt[t] = 0.0f;
}

// ---------------------------------------------------------------------------
// Kernel 1: class_sum[c][f] = sum_{b,v} emb[b,f,v]*(y[b,c,v]>0), count[c]
// Double-buffered async-LDS staging + WMMA f32 16x16x4. Memory-bound (~320MB).
// ---------------------------------------------------------------------------
__global__ __launch_bounds__(128) void class_sum_kernel(
    const float* __restrict__ emb, const int* __restrict__ y,
    float* __restrict__ cs, float* __restrict__ cnt)
{
    __shared__ float sE[2][F_][EPAD];   // 2 x (64 x 68) fp32 emb tiles
    __shared__ int   sY[2][16][EPAD];   // rows 0..3 = raw y, rows 4..15 = 0

    const int tid  = threadIdx.x;
    const int b    = blockIdx.y;
    const int lane = tid & 31;
    const int wave = tid >> 5;          // 0..3 -> f-tile
    const int f0   = wave * 16;
    const int m    = lane & 15;         // A-matrix row (f within tile)
    const int kb   = (lane >> 4) * 2;   // K offset: lanes<16 -> 0, lanes>=16 -> 2
    const int n    = lane & 15;         // B/D column (class for n<4)

    // zero both y buffers once (pad rows 4..15 must stay 0 for B-fragment)
    for (int e = tid; e < 2 * 16 * EPAD; e += 128) ((int*)sY)[e] = 0;
    __syncthreads();

    v8f acc = {};
    int  cntLoc = 0;
    const float* embB = emb + (size_t)b * F_ * V_;
    const int*   yB   = y   + (size_t)b * C_ * V_;

    const int step = gridDim.x;
    int chunk = blockIdx.x;

    // prologue: stage first chunk into buffer 0
    if (chunk < NCHUNK) {
        const int vbase = chunk * VB;
        #pragma unroll
        for (int i = 0; i < 8; ++i) {
            int e = i * 128 + tid;               // 0..1023
            int f = e >> 4, q = e & 15;
            async_copy_b128(&sE[0][f][q * 4], embB + (size_t)f * V_ + vbase + q * 4);
        }
        if (tid < 64) {
            int c = tid >> 4, q = tid & 15;
            async_copy_b128(&sY[0][c][q * 4], yB + (size_t)c * V_ + vbase + q * 4);
        }
    }

    int p = 0;
    for (; chunk < NCHUNK; chunk += step, p ^= 1) {
        wait_async_zero();        // this wave's async writes to buf p done
        __syncthreads();          // all waves' writes done; prior compute on buf p^1 done

        // stage next chunk into the other buffer (no wait -> overlaps compute)
        int nxt = chunk + step;
        if (nxt < NCHUNK) {
            const int vbase = nxt * VB;
            #pragma unroll
            for (int i = 0; i < 8; ++i) {
                int e = i * 128 + tid;
                int f = e >> 4, q = e & 15;
                async_copy_b128(&sE[p ^ 1][f][q * 4], embB + (size_t)f * V_ + vbase + q * 4);
            }
            if (tid < 64) {
                int c = tid >> 4, q = tid & 15;
                async_copy_b128(&sY[p ^ 1][c][q * 4], yB + (size_t)c * V_ + vbase + q * 4);
            }
        }

        // compute from buffer p: 16 WMMA steps, K=4 v's each
        const float (*sEp)[EPAD] = sE[p];
        const int   (*sYp)[EPAD] = sY[p];
        #pragma unroll
        for (int vk = 0; vk < VB; vk += 4) {
            v2f a, bb;
            a.x = sEp[f0 + m][vk + kb];
            a.y = sEp[f0 + m][vk + kb + 1];
            bb.x = sYp[n][vk + kb]     > 0 ? 1.0f : 0.0f;   // rows n>=4 are 0
            bb.y = sYp[n][vk + kb + 1] > 0 ? 1.0f : 0.0f;
            acc = __builtin_amdgcn_wmma_f32_16x16x4_f32(
                false, a, false, bb, (short)0, acc, false, false);
        }
        // per-class positive count from the raw y tile
        if (tid < 64) {
            int c = tid >> 4, q = (tid & 15) * 4;
            cntLoc += (sYp[c][q] > 0) + (sYp[c][q + 1] > 0)
                    + (sYp[c][q + 2] > 0) + (sYp[c][q + 3] > 0);
        }
    }

    // D layout: vgpr r -> M=r (lanes 0-15) / M=8+r (lanes 16-31); N = lane&15
    if (n < 4) {
        int half = (lane >> 4) * 8;
        #pragma unroll
        for (int r = 0; r < 8; ++r) {
            int f = f0 + half + r;
            atomicAdd(&cs[n * F_ + f], acc[r]);
        }
    }
    if (tid < 64) atomicAdd(&cnt[tid >> 4], (float)cntLoc);
}

// ---------------------------------------------------------------------------
// Kernel 2: weights = prod_c proba; per-segment (4096) local top-100.
// Per-thread register-resident elements; incremental argmax (winner rescans).
// ---------------------------------------------------------------------------
__global__ __launch_bounds__(256) void topk_phase1_kernel(
    const float* __restrict__ proba, float* __restrict__ candVal,
    int* __restrict__ candIdx)
{
    __shared__ float wrv[8];
    __shared__ int   wri[8];
    __shared__ int   sBestI;

    int tid = threadIdx.x;
    int seg = blockIdx.x;            // 0..63
    int b   = blockIdx.y;
    int segBase = seg * 4096;
    const float* p = proba + (size_t)b * C_ * V_ + segBase + tid;

    float w[16];
    #pragma unroll
    for (int i = 0; i < 16; ++i) {
        int o = i * 256;
        w[i] = p[o] * p[V_ + o] * p[2 * V_ + o] * p[3 * V_ + o];
    }

    float rv = -2.0f; int ri = tid;
    #pragma unroll
    for (int i = 0; i < 16; ++i)
        if (w[i] > rv) { rv = w[i]; ri = tid + i * 256; }  // ascending -> min idx on tie

    for (int it = 0; it < K_; ++it) {
        float v = rv; int ix = ri;
        #pragma unroll
        for (int off = 16; off > 0; off >>= 1) {
            float ov = __shfl_xor(v, off, 32);
            int   oi = __shfl_xor(ix, off, 32);
            if (ov > v || (ov == v && oi < ix)) { v = ov; ix = oi; }
        }
        if ((tid & 31) == 0) { wrv[tid >> 5] = v; wri[tid >> 5] = ix; }
        __syncthreads();
        if (tid == 0) {
            float bv = wrv[0]; int bi = wri[0];
            for (int wq = 1; wq < 8; ++wq)
                if (wrv[wq] > bv || (wrv[wq] == bv && wri[wq] < bi)) { bv = wrv[wq]; bi = wri[wq]; }
            sBestI = bi;
            candVal[((size_t)b * 64 + seg) * K_ + it] = bv;
            candIdx[((size_t)b * 64 + seg) * K_ + it] = segBase + bi;
        }
        __syncthreads();
        if (ri == sBestI) {                       // unique owner (ri % 256 == tid)
            int slot = (ri - tid) >> 8;
            #pragma unroll
            for (int i = 0; i < 16; ++i) if (i == slot) w[i] = -1.0f;
            rv = -2.0f; ri = tid;
            #pragma unroll
            for (int i = 0; i < 16; ++i)
                if (w[i] > rv) { rv = w[i]; ri = tid + i * 256; }
        }
    }
}

// ---------------------------------------------------------------------------
// Kernel 3: merge 64*100 candidates per batch -> exact global top-100
// (descending; ties -> smaller global index, matching lax.top_k)
// ---------------------------------------------------------------------------
__global__ __launch_bounds__(256) void topk_phase2_kernel(
    const float* __restrict__ candVal, const int* __restrict__ candIdx,
    int* __restrict__ idxOut)
{
    __shared__ float wrv[8];
    __shared__ int   wrg[8];
    __shared__ int   wrp[8];
    __shared__ int   sBestP;

    int tid = threadIdx.x;
    int b   = blockIdx.x;
    const float* cv = candVal + (size_t)b * 6400 + tid;
    const int*   ci = candIdx + (size_t)b * 6400 + tid;

    float w[25]; int g[25];
    #pragma unroll
    for (int i = 0; i < 25; ++i) { w[i] = cv[i * 256]; g[i] = ci[i * 256]; }

    float rv = -2.0f; int rg = 0x7fffffff; int rp = tid;
    #pragma unroll
    for (int i = 0; i < 25; ++i)
        if (w[i] > rv || (w[i] == rv && g[i] < rg)) { rv = w[i]; rg = g[i]; rp = tid + i * 256; }

    for (int it = 0; it < K_; ++it) {
        float v = rv; int gg = rg; int pp = rp;
        #pragma unroll
        for (int off = 16; off > 0; off >>= 1) {
            float ov = __shfl_xor(v, off, 32);
            int   og = __shfl_xor(gg, off, 32);
            int   op = __shfl_xor(pp, off, 32);
            if (ov > v || (ov == v && og < gg)) { v = ov; gg = og; pp = op; }
        }
        if ((tid & 31) == 0) { wrv[tid >> 5] = v; wrg[tid >> 5] = gg; wrp[tid >> 5] = pp; }
        __syncthreads();
        if (tid == 0) {
            float bv = wrv[0]; int bg = wrg[0]; int bp = wrp[0];
            for (int wq = 1; wq < 8; ++wq)
                if (wrv[wq] > bv || (wrv[wq] == bv && wrg[wq] < bg)) { bv = wrv[wq]; bg = wrg[wq]; bp = wrp[wq]; }
            sBestP = bp;
            idxOut[b * K_ + it] = bg;
        }
        __syncthreads();
        if (rp == sBestP) {
            int slot = (rp - tid) >> 8;
            #pragma unroll
            for (int i = 0; i < 25; ++i) if (i == slot) w[i] = -1.0f;
            rv = -2.0f; rg = 0x7fffffff; rp = tid;
            #pragma unroll
            for (int i = 0; i < 25; ++i)
                if (w[i] > rv || (w[i] == rv && g[i] < rg)) { rv = w[i]; rg = g[i]; rp = tid + i * 256; }
        }
    }
}

// ---------------------------------------------------------------------------
// Kernel 4: finalize EMA -> nar, gather he, compute contrastive loss (B,K)
// One 64-thread block per (b,k) item.
// ---------------------------------------------------------------------------
__global__ __launch_bounds__(64) void loss_kernel(
    const float* __restrict__ emb, const int* __restrict__ y,
    const float* __restrict__ avg_repr,
    const float* __restrict__ cs, const float* __restrict__ cnt,
    const int* __restrict__ idxOut, float* __restrict__ out)
{
    __shared__ float sNar[3 * F_];
    __shared__ float part[2];
    __shared__ int   sV;

    int f = threadIdx.x;       // 0..63
    int k = blockIdx.x;        // 0..99
    int b = blockIdx.y;        // 0..3

    #pragma unroll
    for (int c = 1; c < C_; ++c) {
        float cc   = cnt[c];
        float mean = cs[c * F_ + f] / fmaxf(cc, 1.0f);
        float ar   = avg_repr[c * F_ + f];
        sNar[(c - 1) * F_ + f] = (cc > 0.0f) ? (ar * 0.1f + mean * 0.9f) : ar;
    }
    if (f == 0) sV = idxOut[b * K_ + k];
    __syncthreads();

    int v = sV;
    float he = emb[(size_t)b * F_ * V_ + (size_t)f * V_ + v];
    // TAU = 0.1 -> divide == multiply by 10
    float s = __expf(he * sNar[f] * 10.0f)
            + __expf(he * sNar[F_ + f] * 10.0f)
            + __expf(he * sNar[2 * F_ + f] * 10.0f);
    float o = -__logf(s);

    #pragma unroll
    for (int off = 16; off > 0; off >>= 1) o += __shfl_xor(o, off, 32);
    if ((f & 31) == 0) part[f >> 5] = o;
    __syncthreads();

    if (f == 0) {
        float total = part[0] + part[1];
        const int* yb = y + (size_t)b * C_ * V_;
        int best = yb[v]; int hc = 0;          // argmax over c, first max
        #pragma unroll
        for (int c = 1; c < C_; ++c) {
            int yv = yb[(size_t)c * V_ + v];
            if (yv > best) { best = yv; hc = c; }
        }
        out[b * K_ + k] = (hc == 0) ? total : 0.0f;
    }
}

// ---------------------------------------------------------------------------
extern "C" void kernel_launch(void* const* d_in, const int* in_sizes, int n_in,
                              void* d_out, int out_size, void* d_ws, size_t ws_size,
                              hipStream_t stream) {
    const float* proba    = (const float*)d_in[0];   // (B,C,V) f32
    const int*   y        = (const int*)d_in[1];     // (B,C,V) i32
    const float* emb      = (const float*)d_in[2];   // (B,F,V) f32
    const float* avg_repr = (const float*)d_in[3];   // (1,C,F) f32
    float* out = (float*)d_out;                      // (B,K) f32

    // workspace layout (4-byte units)
    float* ws_f    = (float*)d_ws;
    float* cs      = ws_f;                 // 256
    float* cnt     = ws_f + 256;           // 4
    float* candVal = ws_f + 260;           // B*64*100 = 25600
    int*   candIdx = (int*)(ws_f + 260 + 25600);     // 25600
    int*   idxOut  = (int*)(ws_f + 260 + 2 * 25600); // B*K = 400

    init_ws_kernel<<<1, 256, 0, stream>>>(cs, cnt);
    class_sum_kernel<<<dim3(128, B_), 128, 0, stream>>>(emb, y, cs, cnt);
    topk_phase1_kernel<<<dim3(64, B_), 256, 0, stream>>>(proba, candVal, candIdx);
    topk_phase2_kernel<<<B_, 256, 0, stream>>>(candVal, candIdx, idxOut);
    loss_kernel<<<dim3(K_, B_), 64, 0, stream>>>(emb, y, avg_repr, cs, cnt, idxOut, out);
}